// GraphConv_43018392437371
// MI455X (gfx1250) — compile-verified
//
#include <hip/hip_runtime.h>

typedef __attribute__((ext_vector_type(2))) float v2f;
typedef __attribute__((ext_vector_type(8))) float v8f;

#define NN 50000
#define NE 800000
#define DF 128

// ---------------------------------------------------------------------------
// Kernel 1: h = x @ W via V_WMMA_F32_16X16X4_F32 (wave32, 16x16 tiles, K=4)
// Block = 256 threads (8 waves). Each wave computes 16 rows x 128 cols.
// W staged into LDS in per-lane fragment order:
//   lds[((nt*32 + kb)*32 + lane)*2 + j] = W[kb*4 + (lane>>4)*2 + j][nt*16 + (lane&15)]
// so the inner-loop B load is a conflict-free ds_load_b64 at lane*8.
// ---------------------------------------------------------------------------
extern "C" __global__ __launch_bounds__(256)
void gc_gemm_xw(const float* __restrict__ x, const float* __restrict__ w,
                float* __restrict__ h)
{
    __shared__ float lds[DF * DF];   // 16384 floats = 64 KB (of 320 KB/WGP)

    const int t = threadIdx.x;
    // Stage + permute W into LDS (coalesced global reads).
    #pragma unroll 4
    for (int i = 0; i < 64; ++i) {
        int idx  = t + i * 256;          // 0..16383
        int k    = idx >> 7;
        int n    = idx & 127;
        float v  = w[idx];
        int kb   = k >> 2, kk = k & 3;
        int hf   = kk >> 1, j  = kk & 1;
        int nt   = n >> 4, lid = n & 15;
        int lane = hf * 16 + lid;
        lds[((nt * 32 + kb) * 32 + lane) * 2 + j] = v;
    }
    __syncthreads();

    const int lane = t & 31;
    const int wv   = t >> 5;
    const int lid  = lane & 15;
    const int hf   = lane >> 4;

    const int stripe = blockIdx.x * 8 + wv;     // 16-row stripe
    if (stripe * 16 >= NN) return;              // wave-uniform exit
    const int m0 = stripe * 16;

    // Preload all 32 A fragments: a[kb] = x[m0+lid][kb*4 + 2*hf + {0,1}]
    const float* xrow = x + (long)(m0 + lid) * DF + 2 * hf;
    v2f a[32];
    #pragma unroll
    for (int kb = 0; kb < 32; ++kb)
        a[kb] = *(const v2f*)(xrow + kb * 4);

    for (int nt = 0; nt < 8; ++nt) {
        const float* bp = lds + nt * 2048 + lane * 2;
        v8f c = {};
        #pragma unroll
        for (int kb = 0; kb < 32; ++kb) {
            v2f b = *(const v2f*)(bp + kb * 64);
            c = __builtin_amdgcn_wmma_f32_16x16x4_f32(
                    false, a[kb], false, b, (short)0, c, false, false);
        }
        // c[r] -> h[m0 + r + 8*hf][nt*16 + lid]
        float* hp = h + (long)(m0 + 8 * hf) * DF + nt * 16 + lid;
        #pragma unroll
        for (int r = 0; r < 8; ++r)
            hp[(long)r * DF] = c[r];
    }
}

// ---------------------------------------------------------------------------
// Kernel 2: out[i][:] = bias[:]   (accumulator init; bias folded in)
// ---------------------------------------------------------------------------
extern "C" __global__ __launch_bounds__(256)
void gc_init_bias(const float* __restrict__ bias, float* __restrict__ out)
{
    int tid = blockIdx.x * 256 + threadIdx.x;          // over NN*DF/4
    float4 bv = ((const float4*)bias)[tid & 31];       // DF/4 == 32
    ((float4*)out)[tid] = bv;
}

// ---------------------------------------------------------------------------
// Kernel 3: edge scatter. One wave32 per edge; lane owns 4 features.
// h[col] row read as coalesced float4; accumulate with f32 global atomics.
// Working set (h + out = 51 MB) is L2-resident (192 MB).
// ---------------------------------------------------------------------------
extern "C" __global__ __launch_bounds__(256)
void gc_edge_scatter(const float* __restrict__ h, const float* __restrict__ vals,
                     const int* __restrict__ row, const int* __restrict__ col,
                     float* __restrict__ out)
{
    int gw   = (blockIdx.x * 256 + threadIdx.x) >> 5;  // global wave id == edge
    int lane = threadIdx.x & 31;
    if (gw >= NE) return;

    int   r = row[gw];
    int   c = col[gw];
    float v = vals[gw];

    float4 hv = *(const float4*)(h + (long)c * DF + lane * 4);
    float* o  = out + (long)r * DF + lane * 4;

    (void)__hip_atomic_fetch_add(o + 0, v * hv.x, __ATOMIC_RELAXED, __HIP_MEMORY_SCOPE_AGENT);
    (void)__hip_atomic_fetch_add(o + 1, v * hv.y, __ATOMIC_RELAXED, __HIP_MEMORY_SCOPE_AGENT);
    (void)__hip_atomic_fetch_add(o + 2, v * hv.z, __ATOMIC_RELAXED, __HIP_MEMORY_SCOPE_AGENT);
    (void)__hip_atomic_fetch_add(o + 3, v * hv.w, __ATOMIC_RELAXED, __HIP_MEMORY_SCOPE_AGENT);
}

// ---------------------------------------------------------------------------
// Kernel 4: in-place ReLU
// ---------------------------------------------------------------------------
extern "C" __global__ __launch_bounds__(256)
void gc_relu(float* __restrict__ out)
{
    int tid = blockIdx.x * 256 + threadIdx.x;          // over NN*DF/4
    float4 v = ((float4*)out)[tid];
    v.x = fmaxf(v.x, 0.0f);
    v.y = fmaxf(v.y, 0.0f);
    v.z = fmaxf(v.z, 0.0f);
    v.w = fmaxf(v.w, 0.0f);
    ((float4*)out)[tid] = v;
}

// ---------------------------------------------------------------------------
extern "C" void kernel_launch(void* const* d_in, const int* in_sizes, int n_in,
                              void* d_out, int out_size, void* d_ws, size_t ws_size,
                              hipStream_t stream)
{
    const float* x    = (const float*)d_in[0];   // [NN, DF]
    const float* w    = (const float*)d_in[1];   // [DF, DF]
    const float* bias = (const float*)d_in[2];   // [DF]
    const float* vals = (const float*)d_in[3];   // [NE]
    const int*   row  = (const int*)d_in[4];     // [NE]
    const int*   col  = (const int*)d_in[5];     // [NE]
    float*       out  = (float*)d_out;           // [NN, DF]
    float*       h    = (float*)d_ws;            // [NN, DF] scratch (25.6 MB)

    // 1) h = x @ W          (3125 stripes of 16 rows; 8 stripes/block)
    gc_gemm_xw<<<(3125 + 7) / 8, 256, 0, stream>>>(x, w, h);
    // 2) out = bias (broadcast)
    gc_init_bias<<<(NN * DF / 4) / 256, 256, 0, stream>>>(bias, out);
    // 3) out[row[e]] += vals[e] * h[col[e]]   (one wave per edge)
    gc_edge_scatter<<<NE * 32 / 256, 256, 0, stream>>>(h, vals, row, col, out);
    // 4) out = relu(out)
    gc_relu<<<(NN * DF / 4) / 256, 256, 0, stream>>>(out);
}